// ChamferImageLoss_85426899517758
// MI455X (gfx1250) — compile-verified
//
#include <hip/hip_runtime.h>

typedef __attribute__((ext_vector_type(2))) float v2f;
typedef __attribute__((ext_vector_type(8))) float v8f;

#define FXC (600.0f / 640.0f)
#define FYC (600.0f / 480.0f)

// ---------------------------------------------------------------------------
// Kernel 1: projection + norm precompute.
//  X side (i < Mp): pred = persp(input*scale+shift); store pts, -2*pts, ||pt||^2
//  Y side (i < Np): samples copied;                  store pts, -2*pts, ||pt||^2
//  Pad slots (>= real count): pts = 0, norm = +INF (never wins a min).
// ---------------------------------------------------------------------------
__global__ void chamfer_prep_kernel(const float* __restrict__ inp,     // (M,3)
                                    const float* __restrict__ samples, // (N,2)
                                    const float* __restrict__ scale,   // (3,)
                                    const float* __restrict__ shift,   // (3,)
                                    float* __restrict__ X2, float* __restrict__ Xm2,
                                    float* __restrict__ XX,
                                    float* __restrict__ Y2, float* __restrict__ Ym2,
                                    float* __restrict__ YY,
                                    int M, int Mp, int N, int Np) {
  const int i = blockIdx.x * blockDim.x + threadIdx.x;
  const float INF = __uint_as_float(0x7f800000u);

  if (i < Mp) {
    if (i < M) {
      float cx = inp[3 * i + 0] * scale[0] + shift[0];
      float cy = inp[3 * i + 1] * scale[1] + shift[1];
      float cz = inp[3 * i + 2] * scale[2] + shift[2];
      float u = (cx * FXC) / cz;
      float v = (cy * FYC) / cz;
      X2[2 * i + 0] = u;
      X2[2 * i + 1] = v;
      Xm2[2 * i + 0] = -2.0f * u;
      Xm2[2 * i + 1] = -2.0f * v;
      XX[i] = u * u + v * v;
    } else {
      X2[2 * i + 0] = 0.0f;
      X2[2 * i + 1] = 0.0f;
      Xm2[2 * i + 0] = 0.0f;
      Xm2[2 * i + 1] = 0.0f;
      XX[i] = INF;
    }
  }
  if (i < Np) {
    if (i < N) {
      float y0 = samples[2 * i + 0];
      float y1 = samples[2 * i + 1];
      Y2[2 * i + 0] = y0;
      Y2[2 * i + 1] = y1;
      Ym2[2 * i + 0] = -2.0f * y0;
      Ym2[2 * i + 1] = -2.0f * y1;
      YY[i] = y0 * y0 + y1 * y1;
    } else {
      Y2[2 * i + 0] = 0.0f;
      Y2[2 * i + 1] = 0.0f;
      Ym2[2 * i + 0] = 0.0f;
      Ym2[2 * i + 1] = 0.0f;
      YY[i] = INF;
    }
  }
}

// ---------------------------------------------------------------------------
// Kernel 2: one chamfer direction.
// Each wave owns one 16-row tile of A and sweeps all column tiles of B.
// d2 tile = WMMA_F32_16x16x4( A = pts(K=2, zero-padded to 4),
//                             B = -2*pts, C = ||a||^2 + ||b||^2 broadcast ).
// Row-min accumulated per lane, min-tree across the 16-lane halves, sum of
// row-mins (rows >= Mrows excluded) written as one partial per wave.
// ---------------------------------------------------------------------------
__global__ void chamfer_dir_kernel(const float* __restrict__ Apts,  // 2 floats/row
                                   const float* __restrict__ Anorm, // 1 float/row
                                   const float* __restrict__ Bm2,   // 2 floats/col (-2*pt)
                                   const float* __restrict__ Bnorm, // 1 float/col
                                   float* __restrict__ partials,    // 1 float/row-tile
                                   int Mrows, int rowTiles, int colTiles) {
  const int lane = threadIdx.x & 31;
  const int wave = threadIdx.x >> 5;
  const int rowTile = blockIdx.x * (blockDim.x >> 5) + wave;
  if (rowTile >= rowTiles) return;  // wave-uniform: EXEC stays all-ones past here

  const int rowBase = rowTile << 4;
  const int lo = lane & 15;
  const int half = lane >> 4;

  // A tile (16x4 f32): lanes 0-15 carry K=0,1; lanes 16-31 carry K=2,3 == 0
  const int arow = rowBase + lo;
  v2f ap = *(const v2f*)(Apts + 2 * arow);
  v2f a;
  a.x = half ? 0.0f : ap.x;
  a.y = half ? 0.0f : ap.y;

  // per-lane ||a||^2 for C build: row of (v, lane) = rowBase + 8*half + v
  float xv[8];
#pragma unroll
  for (int v = 0; v < 8; ++v) xv[v] = Anorm[rowBase + 8 * half + v];

  const float INF = __uint_as_float(0x7f800000u);
  v8f rmin;
#pragma unroll
  for (int v = 0; v < 8; ++v) rmin[v] = INF;

#pragma unroll 2
  for (int ct = 0; ct < colTiles; ++ct) {
    const int bcol = (ct << 4) + lo;
    v2f bp = *(const v2f*)(Bm2 + 2 * bcol);
    v2f b;
    b.x = half ? 0.0f : bp.x;
    b.y = half ? 0.0f : bp.y;
    const float yn = Bnorm[bcol];  // ||b||^2 of this lane's column

    v8f c;
#pragma unroll
    for (int v = 0; v < 8; ++v) c[v] = xv[v] + yn;

    // d2 tile: D = A*B + C   (emits v_wmma_f32_16x16x4_f32)
    v8f d = __builtin_amdgcn_wmma_f32_16x16x4_f32(
        /*neg_a=*/false, a, /*neg_b=*/false, b,
        /*c_mod=*/(short)0, c, /*reuse_a=*/false, /*reuse_b=*/false);

#pragma unroll
    for (int v = 0; v < 8; ++v) rmin[v] = fminf(rmin[v], d[v]);
  }

  // min across the 16-lane halves (masks 1,2,4,8 stay within each half)
#pragma unroll
  for (int v = 0; v < 8; ++v) {
    float r = rmin[v];
    r = fminf(r, __shfl_xor(r, 1, 32));
    r = fminf(r, __shfl_xor(r, 2, 32));
    r = fminf(r, __shfl_xor(r, 4, 32));
    r = fminf(r, __shfl_xor(r, 8, 32));
    rmin[v] = r;
  }

  // fixed-order sum of the 8 row-mins this half owns (skip pad rows)
  float s = 0.0f;
#pragma unroll
  for (int v = 0; v < 8; ++v) {
    const int row = rowBase + 8 * half + v;
    s += (row < Mrows) ? rmin[v] : 0.0f;
  }
  s += __shfl_xor(s, 16, 32);  // combine the two halves
  if (lane == 0) partials[rowTile] = s;
}

// ---------------------------------------------------------------------------
// Kernel 3: deterministic final reduction -> scalar loss.
// ---------------------------------------------------------------------------
__global__ void chamfer_reduce_kernel(const float* __restrict__ pA, int nA,
                                      const float* __restrict__ pB, int nB,
                                      float invM, float invN,
                                      float* __restrict__ out) {
  __shared__ float sm[256];
  const int t = threadIdx.x;
  float a = 0.0f, b = 0.0f;
  for (int i = t; i < nA; i += 256) a += pA[i];
  for (int i = t; i < nB; i += 256) b += pB[i];
  sm[t] = a * invM + b * invN;
  __syncthreads();
  for (int s = 128; s > 0; s >>= 1) {
    if (t < s) sm[t] += sm[t + s];
    __syncthreads();
  }
  if (t == 0) out[0] = sm[0];
}

// ---------------------------------------------------------------------------
extern "C" void kernel_launch(void* const* d_in, const int* in_sizes, int n_in,
                              void* d_out, int out_size, void* d_ws, size_t ws_size,
                              hipStream_t stream) {
  const float* inp     = (const float*)d_in[0];  // (M,3)
  const float* samples = (const float*)d_in[1];  // (1,N,2)
  const float* scale   = (const float*)d_in[2];  // (3,)
  const float* shift   = (const float*)d_in[3];  // (3,)

  const int M = in_sizes[0] / 3;
  const int N = in_sizes[1] / 2;
  const int Mp = (M + 15) & ~15;
  const int Np = (N + 15) & ~15;
  const int rtA = Mp >> 4;  // row tiles for X->Y direction
  const int rtB = Np >> 4;  // row tiles for Y->X direction

  float* ws  = (float*)d_ws;
  float* X2  = ws;             // 2*Mp : pred points
  float* Xm2 = X2 + 2 * Mp;    // 2*Mp : -2*pred
  float* XX  = Xm2 + 2 * Mp;   // Mp   : ||pred||^2 (+INF pad)
  float* Y2  = XX + Mp;        // 2*Np : samples
  float* Ym2 = Y2 + 2 * Np;    // 2*Np : -2*samples
  float* YY  = Ym2 + 2 * Np;   // Np   : ||sample||^2 (+INF pad)
  float* pA  = YY + Np;        // rtA  : partial row-min sums (p2gt)
  float* pB  = pA + rtA;       // rtB  : partial col-min sums (gt2p)

  const int nPrep = (Mp > Np) ? Mp : Np;
  chamfer_prep_kernel<<<(nPrep + 255) / 256, 256, 0, stream>>>(
      inp, samples, scale, shift, X2, Xm2, XX, Y2, Ym2, YY, M, Mp, N, Np);

  // direction 1: rows = pred points, cols = samples  -> rowmin sums
  chamfer_dir_kernel<<<(rtA + 7) / 8, 256, 0, stream>>>(
      X2, XX, Ym2, YY, pA, M, rtA, Np >> 4);
  // direction 2: rows = samples, cols = pred points  -> colmin sums
  chamfer_dir_kernel<<<(rtB + 7) / 8, 256, 0, stream>>>(
      Y2, YY, Xm2, XX, pB, N, rtB, Mp >> 4);

  chamfer_reduce_kernel<<<1, 256, 0, stream>>>(
      pA, rtA, pB, rtB, 1.0f / (float)M, 1.0f / (float)N, (float*)d_out);
}